// RepeatEncoder_317827580133
// MI455X (gfx1250) — compile-verified
//
#include <hip/hip_runtime.h>
#include <cstdint>

// Problem dims (fixed by the reference): inputs [B,L,C] f32, output [T,B,C,L] f32
#define BB 128
#define LL 168
#define CC 64
#define TT 32

#define LT 32                               // l-tile per block (last tile = 8)
#define TILES_PER_B 6                       // ceil(168/32)
#define ROWSTRIDE (CC + 4)                  // 68 floats = 272B: keeps 16B-aligned rows for b128 LDS writes
#define NTHREADS 256

typedef __attribute__((ext_vector_type(4))) float v4f;

__global__ __launch_bounds__(NTHREADS)
void lif_repeat_encoder(const float* __restrict__ in, float* __restrict__ out) {
    __shared__ float tile[LT * ROWSTRIDE];  // 8704 B

    const int tid = threadIdx.x;
    const int blk = blockIdx.x;
    const int b   = blk / TILES_PER_B;
    const int tb  = blk - b * TILES_PER_B;
    const int l0  = tb * LT;
    const int lrem = (LL - l0 < LT) ? (LL - l0) : LT;   // 32 or 8
    const int nflt = lrem * CC;                          // floats in this tile

    // Tile in[b, l0:l0+lrem, 0:CC] is a CONTIGUOUS chunk of global memory.
    const float* gbase = in + ((size_t)b * LL + l0) * CC;

    // ---- CDNA5 async global -> LDS staging (16B per lane per issue) ----
    for (int f = tid * 4; f < nflt; f += NTHREADS * 4) {
        const int row = f >> 6;                          // / CC
        const int col = f & (CC - 1);
        unsigned lofs = (unsigned)(uintptr_t)(const void*)&tile[row * ROWSTRIDE + col];
        unsigned long long gp = (unsigned long long)(uintptr_t)(const void*)(gbase + f);
        asm volatile("global_load_async_to_lds_b128 %0, %1, off"
                     :: "v"(lofs), "v"(gp)
                     : "memory");
    }
    asm volatile("s_wait_asynccnt 0" ::: "memory");      // wait this wave's async copies
    __syncthreads();                                     // make LDS visible block-wide

    // ---- compute + streaming stores ----
    const int lgn  = lrem >> 2;                          // float4-groups per channel row: 8 or 2
    const int lgsh = (lgn == 8) ? 3 : 1;
    const int ngrp = lgn * CC;                           // 512 or 128
    const size_t tstr = ((size_t)BB * CC * LL) >> 2;     // timestep stride in float4 units

    for (int h = tid; h < ngrp; h += NTHREADS) {
        const int c  = h >> lgsh;                        // channel
        const int lg = h & (lgn - 1);                    // l-group (fast across lanes -> coalesced stores)
        const int lb = lg << 2;

        // transposed LDS reads (bank pattern near conflict-free)
        const float x0 = tile[(lb + 0) * ROWSTRIDE + c];
        const float x1 = tile[(lb + 1) * ROWSTRIDE + c];
        const float x2 = tile[(lb + 2) * ROWSTRIDE + c];
        const float x3 = tile[(lb + 3) * ROWSTRIDE + c];

        float v0 = 0.f, v1 = 0.f, v2 = 0.f, v3 = 0.f;
        v4f* op = (v4f*)(out + ((size_t)b * CC + c) * LL + (size_t)(l0 + lb));

#pragma unroll
        for (int t = 0; t < TT; ++t) {
            // v = v + (x - v)/tau, tau = 2  (mul by 0.5 is exact -> matches reference rounding)
            v0 += (x0 - v0) * 0.5f;
            v1 += (x1 - v1) * 0.5f;
            v2 += (x2 - v2) * 0.5f;
            v3 += (x3 - v3) * 0.5f;
            v4f r;
            r.x = (v0 >= 1.0f) ? 1.0f : 0.0f;
            r.y = (v1 >= 1.0f) ? 1.0f : 0.0f;
            r.z = (v2 >= 1.0f) ? 1.0f : 0.0f;
            r.w = (v3 >= 1.0f) ? 1.0f : 0.0f;
            v0 = (v0 >= 1.0f) ? 0.0f : v0;               // hard reset to v_reset = 0
            v1 = (v1 >= 1.0f) ? 0.0f : v1;
            v2 = (v2 >= 1.0f) ? 0.0f : v2;
            v3 = (v3 >= 1.0f) ? 0.0f : v3;
            __builtin_nontemporal_store(r, op);          // b128 NT store, coalesced across lanes
            op += tstr;
        }
    }
}

extern "C" void kernel_launch(void* const* d_in, const int* in_sizes, int n_in,
                              void* d_out, int out_size, void* d_ws, size_t ws_size,
                              hipStream_t stream) {
    (void)in_sizes; (void)n_in; (void)out_size; (void)d_ws; (void)ws_size;
    const float* in = (const float*)d_in[0];
    float* out = (float*)d_out;
    dim3 grid(BB * TILES_PER_B);   // 768 blocks
    dim3 block(NTHREADS);          // 8 waves (wave32)
    hipLaunchKernelGGL(lif_repeat_encoder, grid, block, 0, stream, in, out);
}